// GCN_5497558139162
// MI455X (gfx1250) — compile-verified
//
#include <hip/hip_runtime.h>
#include <hip/hip_bf16.h>

typedef __attribute__((ext_vector_type(2))) float v2f;
typedef __attribute__((ext_vector_type(8))) float v8f;

#define NNODES 100000
#define NEDGES 1600000
#define INF    128
#define HF     64
#define OUTF   32

// ---------- degree / normalization ----------
__global__ void k_init_deg(float* __restrict__ deg) {
    int i = blockIdx.x * blockDim.x + threadIdx.x;
    if (i < NNODES) deg[i] = 1.0f;   // self-loop weight contributes 1
}

__global__ void k_deg_accum(const long long* __restrict__ col,
                            const float* __restrict__ ew,
                            float* __restrict__ deg) {
    int e = blockIdx.x * blockDim.x + threadIdx.x;
    if (e < NEDGES) atomicAdd(&deg[(int)col[e]], ew[e]);
}

__global__ void k_dinv(float* __restrict__ deg) {
    int i = blockIdx.x * blockDim.x + threadIdx.x;
    if (i < NNODES) {
        float d = deg[i];
        deg[i] = (d > 0.0f) ? rsqrtf(d) : 0.0f;
    }
}

// ---------- WMMA f32 GEMM:  C[M,N] = A[M,K] @ W[K,N] ----------
// One wave -> 16 rows x N cols. K,N multiples of 4/16. M multiple of 16.
template<int K, int N>
__global__ __launch_bounds__(128)
void k_gemm_wmma(const float* __restrict__ A,
                 const float* __restrict__ W,
                 float* __restrict__ C, int M) {
    __shared__ float lw[K * N];
    for (int i = threadIdx.x; i < K * N; i += 128) lw[i] = W[i];
    __syncthreads();

    const int wave = threadIdx.x >> 5;
    const int lane = threadIdx.x & 31;
    const int tile = blockIdx.x * 4 + wave;
    const int row0 = tile * 16;
    if (row0 >= M) return;            // uniform per wave: EXEC stays all-1s below

    const int ml = lane & 15;         // M (for A) / N (for B) lane index
    const int kh = (lane >> 4) << 1;  // K sub-offset: lanes 0-15 -> 0, 16-31 -> 2

    const float* arow = A + (size_t)(row0 + ml) * K;

    constexpr int NT = N / 16;
    v8f acc[NT] = {};

    for (int kk = 0; kk < K; kk += 4) {
        const int kb = kk + kh;
        // A 16x4 f32 layout: lane holds M=ml, VGPR j -> K = kb + j
        v2f a = *(const v2f*)(arow + kb);
#pragma unroll
        for (int t = 0; t < NT; ++t) {
            // B 4x16 f32 layout: VGPR j, lane -> K = kb + j, N = t*16 + ml
            v2f b;
            b.x = lw[(kb    ) * N + t * 16 + ml];
            b.y = lw[(kb + 1) * N + t * 16 + ml];
            acc[t] = __builtin_amdgcn_wmma_f32_16x16x4_f32(
                false, a, false, b, (short)0, acc[t], false, false);
        }
    }

    // C 16x16 f32 layout: VGPR v -> M = v + 8*(lane>=16), N = ml
    const int rb = row0 + ((lane >> 4) << 3);
#pragma unroll
    for (int t = 0; t < NT; ++t) {
#pragma unroll
        for (int v = 0; v < 8; ++v) {
            C[(size_t)(rb + v) * N + t * 16 + ml] = acc[t][v];
        }
    }
}

// ---------- self-loop + bias init:  agg[i,f] = dinv[i]^2 * h[i,f] + bias[f] ----------
template<int F>
__global__ void k_init_agg(const float* __restrict__ dinv,
                           const float* __restrict__ h,
                           const float* __restrict__ bias,
                           float* __restrict__ agg) {
    int t = blockIdx.x * blockDim.x + threadIdx.x;
    if (t >= NNODES * F) return;
    int i = t / F;
    int f = t - i * F;
    float di = dinv[i];
    agg[t] = di * di * h[t] + bias[f];
}

// ---------- edge scatter: wave per edge, coalesced feature row + f32 atomics ----------
template<int F>
__global__ __launch_bounds__(256)
void k_edge_agg(const long long* __restrict__ rows,
                const long long* __restrict__ cols,
                const float* __restrict__ ew,
                const float* __restrict__ dinv,
                const float* __restrict__ h,
                float* __restrict__ agg) {
    int gt = blockIdx.x * blockDim.x + threadIdx.x;
    int e = gt >> 5;
    int lane = gt & 31;
    if (e >= NEDGES) return;
    int r = (int)rows[e];
    int c = (int)cols[e];
    float norm = dinv[r] * ew[e] * dinv[c];
    const float* hr = h + (size_t)r * F;
    float* ar = agg + (size_t)c * F;
#pragma unroll
    for (int i = 0; i < F / 32; ++i) {
        int f = lane + i * 32;
        atomicAdd(&ar[f], norm * hr[f]);
    }
}

__global__ void k_relu(float* __restrict__ x, int n) {
    int i = blockIdx.x * blockDim.x + threadIdx.x;
    if (i < n) x[i] = fmaxf(x[i], 0.0f);
}

extern "C" void kernel_launch(void* const* d_in, const int* in_sizes, int n_in,
                              void* d_out, int out_size, void* d_ws, size_t ws_size,
                              hipStream_t stream) {
    const float*     x   = (const float*)d_in[0];
    const long long* ei  = (const long long*)d_in[1];   // int64 [2, E]
    const float*     ew  = (const float*)d_in[2];
    const float*     W1  = (const float*)d_in[3];
    const float*     b1  = (const float*)d_in[4];
    const float*     W2  = (const float*)d_in[5];
    const float*     b2  = (const float*)d_in[6];
    float*           out = (float*)d_out;

    const long long* row = ei;
    const long long* col = ei + NEDGES;

    float* ws   = (float*)d_ws;
    float* dinv = ws;                          // NNODES
    float* h1   = dinv + NNODES;               // NNODES*HF
    float* agg1 = h1 + (size_t)NNODES * HF;    // NNODES*HF   (becomes h2 after relu)
    float* z    = agg1 + (size_t)NNODES * HF;  // NNODES*OUTF

    const int TB = 256;
    // 1) degree (self-loop = 1) + edge-weight accumulation -> D^-1/2
    k_init_deg<<<(NNODES + TB - 1) / TB, TB, 0, stream>>>(dinv);
    k_deg_accum<<<(NEDGES + TB - 1) / TB, TB, 0, stream>>>(col, ew, dinv);
    k_dinv<<<(NNODES + TB - 1) / TB, TB, 0, stream>>>(dinv);

    // 2) layer 1: h1 = x @ W1 (WMMA f32), agg = selfloop + bias, edge scatter, relu
    {
        int tiles = NNODES / 16;                       // 6250, exact
        k_gemm_wmma<INF, HF><<<(tiles + 3) / 4, 128, 0, stream>>>(x, W1, h1, NNODES);
        int n = NNODES * HF;
        k_init_agg<HF><<<(n + TB - 1) / TB, TB, 0, stream>>>(dinv, h1, b1, agg1);
        long long wthreads = (long long)NEDGES * 32;
        k_edge_agg<HF><<<(int)((wthreads + TB - 1) / TB), TB, 0, stream>>>(row, col, ew, dinv, h1, agg1);
        k_relu<<<(n + TB - 1) / TB, TB, 0, stream>>>(agg1, n);
    }

    // 3) layer 2: z = h2 @ W2 (WMMA f32), out = selfloop + bias, edge scatter
    {
        int tiles = NNODES / 16;
        k_gemm_wmma<HF, OUTF><<<(tiles + 3) / 4, 128, 0, stream>>>(agg1, W2, z, NNODES);
        int n = NNODES * OUTF;
        k_init_agg<OUTF><<<(n + TB - 1) / TB, TB, 0, stream>>>(dinv, z, b2, out);
        long long wthreads = (long long)NEDGES * 32;
        k_edge_agg<OUTF><<<(int)((wthreads + TB - 1) / TB), TB, 0, stream>>>(row, col, ew, dinv, z, out);
    }
}